// MultiHeadAttentionLayer_59614146068507
// MI455X (gfx1250) — compile-verified
//
#include <hip/hip_runtime.h>
#include <hip/hip_bf16.h>
#include <math.h>

// EGNN-style multi-head graph attention for MI455X (gfx1250, wave32).
// Phases:
//   1. zero scratch accumulators (wV, za)
//   2. fused Q/K/V/S GEMM: feat(50000x256) x W^T(256x128) via V_WMMA_F32_16X16X4_F32
//      (software-pipelined fragment loads so waits overlap WMMA issue)
//   3. per-edge scatter: 1 wave32 per edge, shfl_xor head reduction, f32 atomics.
//      z is accumulated FACTORIZED: proj_e is linear in (e0,e1,e2,1), so
//      z[d,h,:] = We . (sum_e s_eh * espcf_e) + be * (sum_e s_eh)  ->
//      only 4 atomics per head per edge (16) instead of 128.
//   4. finalize: reconstruct z from [N,H,4] accumulator, out = S + wV/(z+1e-6)

typedef __attribute__((ext_vector_type(2))) float v2f;
typedef __attribute__((ext_vector_type(8))) float v8f;

#define IN_DIM 256
#define HD 128
#define EPSE 1e-7f

// ---------------------------------------------------------------- zero scratch
__global__ void zero_ws_kernel(float* __restrict__ p, size_t count) {
  size_t i = (size_t)blockIdx.x * blockDim.x + threadIdx.x;
  if (i < count) p[i] = 0.0f;
}

// ------------------------------------------------------- fused Q/K/V/S GEMM
// One wave computes one 16x16 output tile for all four projections.
// A fragment (16x4 f32): lane<16 holds feat[row][k0..k0+1], lane>=16 holds k0+2..k0+3.
// B fragment (4x16 f32): same K-pairing with column index from lane&15.
// C/D layout: vgpr r -> row (r + 8*(lane>>4)), col (lane&15).
__global__ __launch_bounds__(256) void qkvs_gemm_kernel(
    const float* __restrict__ feat,
    const float* __restrict__ Wq, const float* __restrict__ bq,
    const float* __restrict__ Wk, const float* __restrict__ bk,
    const float* __restrict__ Wv, const float* __restrict__ bv,
    const float* __restrict__ Wsw, const float* __restrict__ bs,
    float* __restrict__ Q, float* __restrict__ K,
    float* __restrict__ V, float* __restrict__ S, int n)
{
  const int wavesPerBlock = blockDim.x >> 5;
  const int wave = blockIdx.x * wavesPerBlock + (threadIdx.x >> 5);
  const int lane = threadIdx.x & 31;
  const int mtiles = (n + 15) >> 4;
  const int total = mtiles * (HD / 16);
  if (wave >= total) return;            // wave-uniform: EXEC stays all-ones for WMMA

  const int tm = wave >> 3;             // 128/16 == 8 column tiles
  const int tc = wave & 7;
  const int half = lane >> 4;           // 0 -> K {0,1}, 1 -> K {2,3}
  const int lrow = lane & 15;

  int arow = tm * 16 + lrow;
  if (arow >= n) arow = n - 1;          // clamp (keeps EXEC full); store is guarded
  const int col = tc * 16 + lrow;

  const float* aptr = feat + (size_t)arow * IN_DIM + half * 2;
  const float* pq = Wq  + (size_t)col * IN_DIM + half * 2;
  const float* pk = Wk  + (size_t)col * IN_DIM + half * 2;
  const float* pv = Wv  + (size_t)col * IN_DIM + half * 2;
  const float* ps = Wsw + (size_t)col * IN_DIM + half * 2;

  v8f accQ = {0,0,0,0,0,0,0,0};
  v8f accK = {0,0,0,0,0,0,0,0};
  v8f accV = {0,0,0,0,0,0,0,0};
  v8f accS = {0,0,0,0,0,0,0,0};

  // software pipeline: fragments for k0 live in regs while k0+4 loads are in flight
  v2f a  = *(const v2f*)(aptr);
  v2f wq = *(const v2f*)(pq);
  v2f wk = *(const v2f*)(pk);
  v2f wv = *(const v2f*)(pv);
  v2f ws = *(const v2f*)(ps);

#pragma unroll 4
  for (int k0 = 0; k0 < IN_DIM - 4; k0 += 4) {
    const v2f a_n  = *(const v2f*)(aptr + k0 + 4);
    const v2f wq_n = *(const v2f*)(pq + k0 + 4);
    const v2f wk_n = *(const v2f*)(pk + k0 + 4);
    const v2f wv_n = *(const v2f*)(pv + k0 + 4);
    const v2f ws_n = *(const v2f*)(ps + k0 + 4);
    accQ = __builtin_amdgcn_wmma_f32_16x16x4_f32(false, a, false, wq, (short)0, accQ, false, false);
    accK = __builtin_amdgcn_wmma_f32_16x16x4_f32(false, a, false, wk, (short)0, accK, false, false);
    accV = __builtin_amdgcn_wmma_f32_16x16x4_f32(false, a, false, wv, (short)0, accV, false, false);
    accS = __builtin_amdgcn_wmma_f32_16x16x4_f32(false, a, false, ws, (short)0, accS, false, false);
    a = a_n; wq = wq_n; wk = wk_n; wv = wv_n; ws = ws_n;
  }
  accQ = __builtin_amdgcn_wmma_f32_16x16x4_f32(false, a, false, wq, (short)0, accQ, false, false);
  accK = __builtin_amdgcn_wmma_f32_16x16x4_f32(false, a, false, wk, (short)0, accK, false, false);
  accV = __builtin_amdgcn_wmma_f32_16x16x4_f32(false, a, false, wv, (short)0, accV, false, false);
  accS = __builtin_amdgcn_wmma_f32_16x16x4_f32(false, a, false, ws, (short)0, accS, false, false);

  const float biasQ = bq[col];
  const float biasK = bk[col];
  const float biasV = bv[col];
  const float biasS = bs[col];

#pragma unroll
  for (int r = 0; r < 8; ++r) {
    const int row = tm * 16 + r + half * 8;
    if (row < n) {
      const size_t o = (size_t)row * HD + col;
      Q[o] = accQ[r] + biasQ;
      K[o] = accK[r] + biasK;
      V[o] = accV[r] + biasV;
      S[o] = accS[r] + biasS;
    }
  }
}

// ------------------------------------------------------------ edge scatter
// One wave32 per edge (grid-stride). Lane L owns channels [4L, 4L+4).
// Head h = lane/8; dot over D=32 reduced by shfl_xor 1/2/4 within 8-lane groups.
// za[d*16 + h*4 + j] accumulates per-head (s*e0, s*e1, s*e2, s).
__global__ __launch_bounds__(256) void edge_scatter_kernel(
    const float* __restrict__ pos,
    const float* __restrict__ We, const float* __restrict__ be,
    const float* __restrict__ Q, const float* __restrict__ K,
    const float* __restrict__ V,
    const int* __restrict__ src, const int* __restrict__ dst,
    float* __restrict__ wV, float* __restrict__ za, int e)
{
  const int lane = threadIdx.x & 31;
  const int wid = blockIdx.x * (blockDim.x >> 5) + (threadIdx.x >> 5);
  const int nwaves = gridDim.x * (blockDim.x >> 5);
  const int c = lane << 2;
  const int j = lane & 7;               // lane index inside head group
  const int head = lane >> 3;

  // Hoist per-channel spatial projection weights: We is [128,3] row-major,
  // rows c..c+3 are 12 consecutive floats, 16B-aligned (lane*48 bytes).
  const float4 w0 = *(const float4*)(We + (size_t)c * 3);
  const float4 w1 = *(const float4*)(We + (size_t)c * 3 + 4);
  const float4 w2 = *(const float4*)(We + (size_t)c * 3 + 8);
  const float4 b4 = *(const float4*)(be + c);
  const float inv_sqrt_d = 0.17677669529663687f;  // 1/sqrt(32)

  for (int ed = wid; ed < e; ed += nwaves) {
    const int sN = src[ed];
    const int dN = dst[ed];

    // prefetch next iteration's gather rows (random-access -> pull toward L0)
    const int ed2 = ed + nwaves;
    if (ed2 < e) {
      const int sN2 = src[ed2];
      const int dN2 = dst[ed2];
      __builtin_prefetch(K + (size_t)sN2 * HD + c, 0, 3);
      __builtin_prefetch(Q + (size_t)dN2 * HD + c, 0, 3);
      __builtin_prefetch(V + (size_t)sN2 * HD + c, 0, 3);
    }

    const float r0 = pos[dN * 3 + 0] - pos[sN * 3 + 0];
    const float r1 = pos[dN * 3 + 1] - pos[sN * 3 + 1];
    const float r2 = pos[dN * 3 + 2] - pos[sN * 3 + 2];
    const float inv = 1.0f / (sqrtf(r0 * r0 + r1 * r1 + r2 * r2) + EPSE);
    const float e0 = (r0 + 1.0f) * inv;
    const float e1 = (r1 + 1.0f) * inv;
    const float e2 = (r2 + 1.0f) * inv;

    // proj_e for this lane's 4 channels
    const float p0 = w0.x * e0 + w0.y * e1 + w0.z * e2 + b4.x;
    const float p1 = w0.w * e0 + w1.x * e1 + w1.y * e2 + b4.y;
    const float p2 = w1.z * e0 + w1.w * e1 + w2.x * e2 + b4.z;
    const float p3 = w2.y * e0 + w2.z * e1 + w2.w * e2 + b4.w;

    const float4 kq = *(const float4*)(K + (size_t)sN * HD + c);
    const float4 qq = *(const float4*)(Q + (size_t)dN * HD + c);
    const float4 vq = *(const float4*)(V + (size_t)sN * HD + c);

    float part = kq.x * qq.x + kq.y * qq.y + kq.z * qq.z + kq.w * qq.w;
    part += __shfl_xor(part, 1);
    part += __shfl_xor(part, 2);
    part += __shfl_xor(part, 4);     // per-head dot(K,Q) across 8 lanes
    const float sc = part * inv_sqrt_d;

    // wV: full per-channel accumulation (does not factorize)
    const float s0 = sc * p0, s1 = sc * p1, s2 = sc * p2, s3 = sc * p3;
    float* wvp = wV + (size_t)dN * HD + c;
    __hip_atomic_fetch_add(wvp + 0, vq.x * s0, __ATOMIC_RELAXED, __HIP_MEMORY_SCOPE_AGENT);
    __hip_atomic_fetch_add(wvp + 1, vq.y * s1, __ATOMIC_RELAXED, __HIP_MEMORY_SCOPE_AGENT);
    __hip_atomic_fetch_add(wvp + 2, vq.z * s2, __ATOMIC_RELAXED, __HIP_MEMORY_SCOPE_AGENT);
    __hip_atomic_fetch_add(wvp + 3, vq.w * s3, __ATOMIC_RELAXED, __HIP_MEMORY_SCOPE_AGENT);

    // z: factorized — lanes j=0..3 of each head group write (s*e0, s*e1, s*e2, s)
    if (j < 4) {
      const float sel = (j == 0) ? e0 : (j == 1) ? e1 : (j == 2) ? e2 : 1.0f;
      __hip_atomic_fetch_add(za + (size_t)dN * 16 + head * 4 + j, sc * sel,
                             __ATOMIC_RELAXED, __HIP_MEMORY_SCOPE_AGENT);
    }
  }
}

// ---------------------------------------------------------------- finalize
// Reconstruct z[d,c] = We[c].(za[d,h,0..2]) + be[c]*za[d,h,3], h = c/32,
// then out = S + wV / (z + 1e-6).
__global__ void finalize_kernel(const float* __restrict__ S,
                                const float* __restrict__ wV,
                                const float* __restrict__ za,
                                const float* __restrict__ We,
                                const float* __restrict__ be,
                                float* __restrict__ out, size_t count) {
  size_t i = (size_t)blockIdx.x * blockDim.x + threadIdx.x;
  if (i < count) {
    const size_t d = i >> 7;            // node
    const int cch = (int)(i & 127);     // channel
    const int h = cch >> 5;             // head
    const float4 zb = *(const float4*)(za + d * 16 + h * 4);
    const float zc = We[cch * 3 + 0] * zb.x + We[cch * 3 + 1] * zb.y +
                     We[cch * 3 + 2] * zb.z + be[cch] * zb.w;
    out[i] = S[i] + wV[i] / (zc + 1e-6f);
  }
}

extern "C" void kernel_launch(void* const* d_in, const int* in_sizes, int n_in,
                              void* d_out, int out_size, void* d_ws, size_t ws_size,
                              hipStream_t stream) {
  const float* feat = (const float*)d_in[0];
  const float* position = (const float*)d_in[1];
  const float* Wq = (const float*)d_in[2];
  const float* bq = (const float*)d_in[3];
  const float* Wk = (const float*)d_in[4];
  const float* bk = (const float*)d_in[5];
  const float* Wv = (const float*)d_in[6];
  const float* bv = (const float*)d_in[7];
  const float* We = (const float*)d_in[8];
  const float* be = (const float*)d_in[9];
  const float* Ws = (const float*)d_in[10];
  const float* bs = (const float*)d_in[11];
  const int* src = (const int*)d_in[12];
  const int* dst = (const int*)d_in[13];

  const int n = in_sizes[0] / IN_DIM;   // 50000
  const int e = in_sizes[12];           // 800000
  const size_t nb = (size_t)n * HD;

  // workspace layout: Q | K | V | S | wV (nb each) | za (n*16)
  float* Q  = (float*)d_ws;
  float* K  = Q + nb;
  float* V  = K + nb;
  float* S  = V + nb;
  float* wV = S + nb;
  float* za = wV + nb;

  {  // zero the scatter accumulators every call (harness poisons d_ws)
    const size_t zc = nb + (size_t)n * 16;
    const int blocks = (int)((zc + 255) / 256);
    zero_ws_kernel<<<blocks, 256, 0, stream>>>(wV, zc);
  }
  {  // fused WMMA GEMM: 8 waves/block, 1 tile/wave
    const int mtiles = (n + 15) / 16;
    const int total = mtiles * (HD / 16);
    const int blocks = (total + 7) / 8;
    qkvs_gemm_kernel<<<blocks, 256, 0, stream>>>(feat, Wq, bq, Wk, bk, Wv, bv,
                                                 Ws, bs, Q, K, V, S, n);
  }
  {  // edge scatter: 32768 waves grid-striding over 800K edges
    edge_scatter_kernel<<<4096, 256, 0, stream>>>(position, We, be, Q, K, V,
                                                  src, dst, wV, za, e);
  }
  {
    const int blocks = (int)((nb + 255) / 256);
    finalize_kernel<<<blocks, 256, 0, stream>>>(S, wV, za, We, be,
                                                (float*)d_out, nb);
  }
}